// LU_45853070852239
// MI455X (gfx1250) — compile-verified
//
#include <hip/hip_runtime.h>
#include <cstdint>

// ---------------------------------------------------------------------------
// Batched 5x5 block-LU pipeline (B=262144 elements, 9 blocks of 5x5 fp32 each).
// Memory-bound (~472 MB total traffic -> ~20us at 23.3 TB/s). Strategy:
//   * async global->LDS streaming (CDNA5 GLOBAL_LOAD_ASYNC_TO_LDS_B128),
//     perfectly coalesced 16B/lane chunks, non-temporal (streamed once, >L2),
//   * per-lane scalar LU from LDS (row stride 225 words == 33 mod 64, odd ->
//     conflict-free across a wave32),
//   * async LDS->global streaming store (GLOBAL_STORE_ASYNC_FROM_LDS_B128, NT).
// ---------------------------------------------------------------------------

#define NPE   225                       // floats per element (9*5*5)
#define TILE  64                        // elements per workgroup (2 waves)
#define CHUNKS ((TILE * NPE * 4) / 16)  // 3600 16-byte chunks per tile

#define AS1 __attribute__((address_space(1)))
#define AS3 __attribute__((address_space(3)))

#define CPOL_TH_NT 1                    // temporal hint: non-temporal (ISA TH=NT)

typedef int v4i __attribute__((vector_size(16)));

#if defined(__HIP_DEVICE_COMPILE__) && \
    __has_builtin(__builtin_amdgcn_global_load_async_to_lds_b128) && \
    __has_builtin(__builtin_amdgcn_global_store_async_from_lds_b128)
#define USE_ASYNC 1
#else
#define USE_ASYNC 0
#endif

#if defined(__HIP_DEVICE_COMPILE__)
static_assert(USE_ASYNC == 1, "gfx1250 async global<->LDS builtins missing");
#endif

__device__ __forceinline__ void wait_async0() {
#if __has_builtin(__builtin_amdgcn_s_wait_asynccnt)
  __builtin_amdgcn_s_wait_asynccnt(0);
#else
  asm volatile("s_wait_asynccnt 0" ::: "memory");
#endif
}

// Unpivoted Doolittle LU on a 5x5 held in registers (fully unrolled).
// Matches reference: col = A[k+1:,k]/piv; A[k+1:,k+1:] -= col * row.
__device__ __forceinline__ void lu5(float* a) {
#pragma unroll
  for (int k = 0; k < 5; ++k) {
    const float inv = 1.0f / a[k * 5 + k];
#pragma unroll
    for (int r = k + 1; r < 5; ++r) a[r * 5 + k] *= inv;
#pragma unroll
    for (int r = k + 1; r < 5; ++r) {
      const float m = a[r * 5 + k];
#pragma unroll
      for (int c = k + 1; c < 5; ++c) a[r * 5 + c] -= m * a[k * 5 + c];
    }
  }
}

__device__ __forceinline__ void ldb(const float* row, int b, float* a) {
#pragma unroll
  for (int i = 0; i < 25; ++i) a[i] = row[b * 25 + i];
}
__device__ __forceinline__ void stb(float* row, int b, const float* a) {
#pragma unroll
  for (int i = 0; i < 25; ++i) row[b * 25 + i] = a[i];
}

__global__ __launch_bounds__(TILE) void lu_pipeline_kernel(
    const float* __restrict__ x, float* __restrict__ out, int nElem) {
  __shared__ float tile[TILE * NPE];   // 57,600 B
  const int tid = threadIdx.x;
  const int blockStart = (int)blockIdx.x * TILE;
  const int remaining = nElem - blockStart;
  const int rows = remaining < TILE ? remaining : TILE;
  const size_t baseF = (size_t)blockStart * NPE;

  // ---- stage tile into LDS (coalesced streaming) ----
  if (rows == TILE) {
#if USE_ASYNC
    AS1 char* gsrc = (AS1 char*)(uintptr_t)(const void*)(x + baseF);
    AS3 char* lb = (AS3 char*)(uint32_t)(uintptr_t)(void*)tile;
    for (int c = tid; c < CHUNKS; c += TILE) {
      __builtin_amdgcn_global_load_async_to_lds_b128(
          (AS1 v4i*)(gsrc + (size_t)c * 16), (AS3 v4i*)(lb + c * 16),
          0, CPOL_TH_NT);
    }
    wait_async0();
#else
    const float4* gs = (const float4*)(x + baseF);
    float4* ls = (float4*)tile;
    for (int c = tid; c < CHUNKS; c += TILE) ls[c] = gs[c];
#endif
  } else {
    for (int i = tid; i < rows * NPE; i += TILE) tile[i] = x[baseF + i];
  }
  __syncthreads();

  // ---- per-lane compute: one element per thread ----
  if (tid < rows) {
    float* row = tile + tid * NPE;
    float a[25];

    // Stage 1: LU blocks 0,1,2,5,6 (block 4 untouched).
    ldb(row, 0, a); lu5(a); stb(row, 0, a);
    const float s0_11 = a[6];
    ldb(row, 1, a); lu5(a); stb(row, 1, a);
    const float s1_11 = a[6];
    ldb(row, 2, a); lu5(a); stb(row, 2, a);
    const float s2_22 = a[12], s2_23 = a[13], s2_32 = a[17], s2_33 = a[18];
    ldb(row, 5, a); lu5(a); stb(row, 5, a);
    const float s5_11 = a[6];
    ldb(row, 6, a); lu5(a); stb(row, 6, a);
    const float s6_33 = a[18], s6_34 = a[19], s6_43 = a[23], s6_44 = a[24];

    // Stage 2: corrections into blocks 3,7; then LU them.
    ldb(row, 3, a);
    a[0] -= s1_11 + s2_22;   // A3[0,0] -= A1[1,1] + A2[2,2]
    a[1] -= s2_23;           // A3[0,1] -= A2[2,3]
    a[5] -= s2_32;           // A3[1,0] -= A2[3,2]
    a[6] -= s2_33;           // A3[1,1] -= A2[3,3]
    lu5(a); stb(row, 3, a);
    const float t3_11 = a[6];

    ldb(row, 7, a);
    a[0] -= s5_11 + s6_33;   // A7[0,0] -= A5[1,1] + A6[3,3]
    a[1] -= s6_34;           // A7[0,1] -= A6[3,4]
    a[5] -= s6_43;           // A7[1,0] -= A6[4,3]
    a[6] -= s6_44;           // A7[1,1] -= A6[4,4]
    lu5(a); stb(row, 7, a);
    const float t7_11 = a[6];

    // Stage 3: correction into block 8, then LU.
    ldb(row, 8, a);
    a[0] -= s0_11 + t3_11 + t7_11;  // A8[0,0] -= A0[1,1]+A3[1,1]+A7[1,1]
    lu5(a); stb(row, 8, a);
  }
  __syncthreads();

  // ---- stream tile back to global ----
  if (rows == TILE) {
#if USE_ASYNC
    AS1 char* gdst = (AS1 char*)(uintptr_t)(void*)(out + baseF);
    AS3 char* lb = (AS3 char*)(uint32_t)(uintptr_t)(void*)tile;
    for (int c = tid; c < CHUNKS; c += TILE) {
      __builtin_amdgcn_global_store_async_from_lds_b128(
          (AS1 v4i*)(gdst + (size_t)c * 16), (AS3 v4i*)(lb + c * 16),
          0, CPOL_TH_NT);
    }
    wait_async0();
#else
    const float4* ls = (const float4*)tile;
    float4* gd = (float4*)(out + baseF);
    for (int c = tid; c < CHUNKS; c += TILE) gd[c] = ls[c];
#endif
  } else {
    for (int i = tid; i < rows * NPE; i += TILE) out[baseF + i] = tile[i];
  }
}

extern "C" void kernel_launch(void* const* d_in, const int* in_sizes, int n_in,
                              void* d_out, int out_size, void* d_ws, size_t ws_size,
                              hipStream_t stream) {
  (void)n_in; (void)out_size; (void)d_ws; (void)ws_size;
  const float* x = (const float*)d_in[0];
  float* out = (float*)d_out;
  const int nElem = in_sizes[0] / NPE;          // 262144 batch elements
  const int grid = (nElem + TILE - 1) / TILE;   // 4096 workgroups
  lu_pipeline_kernel<<<grid, TILE, 0, stream>>>(x, out, nElem);
}